// ClassifierCaps_74878459838516
// MI455X (gfx1250) — compile-verified
//
#include <hip/hip_runtime.h>
#include <math.h>

// ---------------------------------------------------------------------------
// Fused CapsNet dynamic routing for MI455X (gfx1250), wave32, WMMA f32.
//
//  u_hat ([10,256,1152,16] = 188MB) is NEVER materialized. Each workgroup
//  owns one (k, 16-batch tile) and runs all 3 routing iterations against
//  L2-resident x (9.4MB) and W (5.9MB) with V_WMMA_F32_16X16X4_F32:
//
//   pass_S:  s[b,d]  = sum_{nc} (c[n,b]*x[b,n,c]) * Wflat[nc,d]
//            -> 16x16x9216 GEMM; K split over 8 waves, 2 acc chains/wave,
//               branch-free (iter 0 uses cbuf pre-filled with 1/N).
//   pass_A:  a[n,b]  = sum_c x[b,n,c] * (Wflat @ v^T)[nc,b]
//            -> Z GEMM [9216x16]*[16x16] in 16-row subtiles; each lane's
//               C fragment holds exactly the 8 c-values of one (n,b).
//
//  Assumed gfx1250 f32 WMMA fragment layout (ISA 7.12.2):
//   A (16x4, 2 VGPR):  lane l: row m=l&15, h=l>>4 -> {A[m,2h], A[m,2h+1]}
//   B (4x16, 2 VGPR):  lane l: col n=l&15        -> {B[2h,n], B[2h+1,n]}
//   C/D (16x16, 8 VGPR): vgpr i, lane l -> C[i + 8*(l>>4)][l&15]
// ---------------------------------------------------------------------------

typedef __attribute__((ext_vector_type(2))) float v2f;
typedef __attribute__((ext_vector_type(8))) float v8f;

#define N_CAPS  1152
#define CIN     8
#define COUT    16
#define BATCH   256
#define KCLS    10
#define BT      16                 // batch tile per workgroup
#define NCF     (N_CAPS * CIN)     // 9216 rows of Wflat
#define THREADS 256
#define WAVES   8

// LDS layout (in floats)
#define OFF_LOGITS 0                               // [1152][16] routing logits b
#define OFF_C      (OFF_LOGITS + N_CAPS * BT)      // [1152][16] softmax coeffs
#define OFF_RED    (OFF_C + N_CAPS * BT)           // [8][256] cross-wave reduce
#define OFF_SV     (OFF_RED + WAVES * 256)         // [16][16] s then v
#define OFF_MX     (OFF_SV + 256)                  // [16] rowmax
#define OFF_ISUM   (OFF_MX + 16)                   // [16] 1/sum
#define LDS_FLOATS (OFF_ISUM + 16)                 // = 39200 floats = 156.8 KB

__device__ __forceinline__ v8f wmma4(v2f a, v2f b, v8f c) {
  // D = A(16x4,f32) * B(4x16,f32) + C ; 8-arg form:
  // (neg_a, A, neg_b, B, c_mod, C, reuse_a, reuse_b)
  return __builtin_amdgcn_wmma_f32_16x16x4_f32(false, a, false, b,
                                               (short)0, c, false, false);
}

__global__ __launch_bounds__(THREADS)
void caps_routing_kernel(const float* __restrict__ x,
                         const float* __restrict__ W,
                         float* __restrict__ out)
{
  extern __shared__ float smem[];
  float* logits = smem + OFF_LOGITS;
  float* cbuf   = smem + OFF_C;
  float* red    = smem + OFF_RED;
  float* sv     = smem + OFF_SV;
  float* mx     = smem + OFF_MX;
  float* isum   = smem + OFF_ISUM;

  const int tid   = threadIdx.x;
  const int lane  = tid & 31;
  const int wave  = tid >> 5;
  const int bl    = lane & 15;     // fragment column / local row index
  const int h     = lane >> 4;     // lane half selects K pair / C row half
  const int btile = blockIdx.x;    // 16 batch-tiles
  const int k     = blockIdx.y;    // 10 classes

  const int b_g = btile * BT + bl;                       // global batch row
  const float* Wk = W + (size_t)k * NCF * COUT;          // Wflat [9216][16]

  // init: cbuf = 1/N (iteration-0 routing coeffs), logits = 0 (accumulate-only)
  for (int i = tid; i < N_CAPS * BT; i += THREADS) {
    cbuf[i]   = 1.0f / (float)N_CAPS;
    logits[i] = 0.0f;
  }
  __syncthreads();

  for (int it = 0; it < 3; ++it) {
    // ================= pass_S : s = (coef .* x-rows) @ Wflat ===============
    // Wave owns K-range [wave*1152, wave*1152+1152); two interleaved chains
    // at nc offsets +2h and +4+2h: c is loop-invariant, n advances by 1/step,
    // so x/W/cbuf are pure strided streams (pointer bumps only).
    v8f acc0 = {0.f,0.f,0.f,0.f,0.f,0.f,0.f,0.f};
    v8f acc1 = {0.f,0.f,0.f,0.f,0.f,0.f,0.f,0.f};
    {
      const int nc_lo = wave * (NCF / WAVES);
      const int n0    = nc_lo >> 3;            // starting n for this wave
      const int c0    = 2 * h;                 // chain-0 c (invariant)
      const int c1    = 4 + 2 * h;             // chain-1 c (invariant)
      const float* xr  = x  + ((size_t)b_g * N_CAPS + n0) * CIN;      // +8/step
      const float* wp0 = Wk + (size_t)(nc_lo + c0) * COUT + bl;       // +128/step
      const float* wp1 = Wk + (size_t)(nc_lo + c1) * COUT + bl;       // +128/step
      const float* cp  = cbuf + n0 * BT + bl;                          // +16/step
      #pragma unroll 4
      for (int i = 0; i < N_CAPS / WAVES; ++i) {   // 144 steps, 2 WMMA each
        const float coef = cp[0];                  // one DS read, both chains
        v2f a0; a0.x = xr[c0]     * coef; a0.y = xr[c0 + 1] * coef;
        v2f b0; b0.x = wp0[0];            b0.y = wp0[COUT];
        acc0 = wmma4(a0, b0, acc0);
        v2f a1; a1.x = xr[c1]     * coef; a1.y = xr[c1 + 1] * coef;
        v2f b1; b1.x = wp1[0];            b1.y = wp1[COUT];
        acc1 = wmma4(a1, b1, acc1);
        xr += CIN; wp0 += 8 * COUT; wp1 += 8 * COUT; cp += BT;
      }
    }
    acc0 += acc1;
    #pragma unroll
    for (int i = 0; i < 8; ++i) red[wave * 256 + i * 32 + lane] = acc0[i];
    __syncthreads();
    {   // cross-wave partial-C reduction -> s[b][d] in sv
      float vsum = 0.f;
      #pragma unroll
      for (int w = 0; w < WAVES; ++w) vsum += red[w * 256 + tid];
      const int i = tid >> 5, l = tid & 31;
      const int row = i + 8 * (l >> 4), col = l & 15;
      sv[row * 16 + col] = vsum;
    }
    __syncthreads();
    // squash per row b (in place: sv becomes v)
    if (tid < 16) {
      float nrm2 = 0.f;
      #pragma unroll
      for (int d = 0; d < COUT; ++d) { float t = sv[tid*16+d]; nrm2 += t*t; }
      const float scale = nrm2 / ((1.f + nrm2) * sqrtf(nrm2 + 1e-30f));
      #pragma unroll
      for (int d = 0; d < COUT; ++d) sv[tid*16+d] *= scale;
    }
    __syncthreads();

    if (it == 2) break;

    // ========== pass_A : logits[n,b] += sum_d u_hat[b,n,d]*v[b,d] ==========
    // B operand = v^T, loaded once: B[row=d, col=b] = v[b][d]
    v2f vB[4];
    #pragma unroll
    for (int j = 0; j < 4; ++j) {
      const int d = 4 * j + 2 * h;
      vB[j].x = sv[bl * 16 + d];
      vB[j].y = sv[bl * 16 + d + 1];
    }
    {
      const float* wrow = Wk + (size_t)(wave * 16 + bl) * COUT;  // +2048/step
      const float* xrow = x + ((size_t)b_g * N_CAPS + 2 * wave + h) * CIN; // +128/step
      float* lg = logits + (2 * wave + h) * BT + bl;             // +256/step
      #pragma unroll 2
      for (int t = 0; t < (NCF / 16) / WAVES; ++t) {   // 72 subtiles/wave
        v8f z = {0.f,0.f,0.f,0.f,0.f,0.f,0.f,0.f};
        #pragma unroll
        for (int j = 0; j < 4; ++j) {                  // K=16 (d) chain
          v2f a; a.x = wrow[4*j + 2*h]; a.y = wrow[4*j + 2*h + 1];
          z = wmma4(a, vB[j], z);
        }
        // lane's C column holds Z[nc=16t'+8h+i][b=bl] : the 8 c's of n=2t'+h
        float aval = 0.f;
        #pragma unroll
        for (int i = 0; i < 8; ++i) aval += xrow[i] * z[i];
        lg[0] += aval;
        wrow += 16 * WAVES * COUT; xrow += 2 * WAVES * CIN; lg += 2 * WAVES * BT;
      }
    }
    __syncthreads();

    // ====== softmax over n for each b: mx, 1/sum, then cbuf = c[n,b] =======
    {
      const int b = tid & 15, grp = tid >> 4;
      float m = -3.4e38f;
      for (int n = grp; n < N_CAPS; n += 16) m = fmaxf(m, logits[n*BT + b]);
      red[tid] = m;
      __syncthreads();
      if (tid < 16) {
        float mm = -3.4e38f;
        for (int g = 0; g < 16; ++g) mm = fmaxf(mm, red[g * 16 + tid]);
        mx[tid] = mm;
      }
      __syncthreads();
      float s = 0.f;
      for (int n = grp; n < N_CAPS; n += 16)
        s += __expf(logits[n * BT + b] - mx[b]);
      red[tid] = s;
      __syncthreads();
      if (tid < 16) {
        float ss = 0.f;
        for (int g = 0; g < 16; ++g) ss += red[g * 16 + tid];
        isum[tid] = 1.f / ss;
      }
      __syncthreads();
      for (int n = grp; n < N_CAPS; n += 16)
        cbuf[n * BT + b] = __expf(logits[n * BT + b] - mx[b]) * isum[b];
      __syncthreads();
    }
  }

  // ================= write v : out[(k*256 + b)*16 + d] =====================
  {
    const int b = tid >> 4, d = tid & 15;
    out[((size_t)k * BATCH + (size_t)btile * BT + b) * COUT + d] = sv[b*16 + d];
  }
}

extern "C" void kernel_launch(void* const* d_in, const int* in_sizes, int n_in,
                              void* d_out, int out_size, void* d_ws, size_t ws_size,
                              hipStream_t stream) {
  const float* x = (const float*)d_in[0];   // [256,1152,8] f32
  const float* W = (const float*)d_in[1];   // [10,1152,8,16] f32
  float* out = (float*)d_out;               // [10,256,1,1,16] f32
  (void)in_sizes; (void)n_in; (void)out_size; (void)d_ws; (void)ws_size;

  dim3 grid(BATCH / BT, KCLS);              // 16 x 10 = 160 workgroups
  size_t shmem = (size_t)LDS_FLOATS * sizeof(float);  // ~157 KB dynamic LDS
  caps_routing_kernel<<<grid, dim3(THREADS), shmem, stream>>>(x, W, out);
}